// KLDiv_BCE_65575560675796
// MI455X (gfx1250) — compile-verified
//
#include <hip/hip_runtime.h>
#include <hip/hip_bf16.h>

// Problem constants (from reference)
#define NN    128
#define CC    7
#define HH    96
#define WW    192
#define HW    (HH*WW)          // 18432
#define CHW   (CC*HW)          // 129024
#define NPTS  100
#define NP    112              // NPTS padded to multiple of 16 (7 tiles)
#define KK    (NN*CC)          // 896
#define BG_N  (NN*HW)          // 2359296
#define HW4   (HW/4)           // 4608
#define CHW4  (CHW/4)          // 32256
#define BG_V  (BG_N/4)         // 589824 float4 elements

// Workspace layout (float offsets)
#define WS_ACC  0              // [0]=bg sum, [1]=kl sum, [2]=kl count
#define WS_P    16             // padded P (k-major): KK x NP
#define WS_G    (WS_P + KK*NP) // padded G (k-major): KK x NP
#define WS_T    (WS_G + KK*NP) // term1: NP
#define WS_M    (WS_T + NP)    // gram matrix: NP x NP

typedef __attribute__((ext_vector_type(2))) float v2f;
typedef __attribute__((ext_vector_type(8))) float v8f;

__global__ void k_init(float* ws) {
    if (threadIdx.x < 3) ws[threadIdx.x] = 0.0f;
}

__device__ __forceinline__ float bce_term(float xv, float tv) {
    return fmaxf(xv, 0.0f) - xv * tv + log1pf(expf(-fabsf(xv)));
}

// BCE-with-logits over channel 0: mean(max(x,0) - x*t + log1p(exp(-|x|)))
// Vectorized float4 (b128) loads: channel-0 rows are HW=18432 contiguous
// floats per image; CHW divisible by 4 keeps 16B alignment per image.
__global__ void k_bg(const float4* __restrict__ x4, const float4* __restrict__ t4,
                     float* __restrict__ ws) {
    __shared__ float sm[256];
    float acc = 0.0f;
    for (int v = blockIdx.x * blockDim.x + threadIdx.x; v < BG_V;
         v += gridDim.x * blockDim.x) {
        int n  = v / HW4;
        int r4 = v - n * HW4;
        int idx = n * CHW4 + r4;               // channel 0, float4 units
        float4 xv = x4[idx];
        float4 tv = t4[idx];
        acc += bce_term(xv.x, tv.x);
        acc += bce_term(xv.y, tv.y);
        acc += bce_term(xv.z, tv.z);
        acc += bce_term(xv.w, tv.w);
    }
    sm[threadIdx.x] = acc;
    __syncthreads();
    for (int s = 128; s > 0; s >>= 1) {
        if ((int)threadIdx.x < s) sm[threadIdx.x] += sm[threadIdx.x + s];
        __syncthreads();
    }
    if (threadIdx.x == 0) atomicAdd(&ws[0], sm[0]);
}

// Gather P,G at (py,px) points into k-major padded layout (KK x NP), zero pad.
__global__ void k_gather(const float* __restrict__ in, const float* __restrict__ tg,
                         const int* __restrict__ px, const int* __restrict__ py,
                         float* __restrict__ ws) {
    int e = blockIdx.x * blockDim.x + threadIdx.x;
    if (e >= KK * NP) return;
    int k = e / NP;
    int i = e - k * NP;
    float p = 0.0f, g = 0.0f;
    if (i < NPTS) {
        int n = k / CC;
        int c = k - n * CC;
        int idx = n * CHW + c * HW + py[i] * WW + px[i];
        p = in[idx];
        g = tg[idx];
    }
    ws[WS_P + k * NP + i] = p;
    ws[WS_G + k * NP + i] = g;
}

// term1[i] = sum_k xlogy(P[k,i], P[k,i])
__global__ void k_term1(float* __restrict__ ws) {
    int i = threadIdx.x;
    if (i >= NP) return;
    float s = 0.0f;
    if (i < NPTS) {
        for (int k = 0; k < KK; ++k) {
            float p = ws[WS_P + k * NP + i];
            s += (p > 0.0f) ? p * logf(p) : 0.0f;
        }
    }
    ws[WS_T + i] = s;
}

// Gram matrix cross[i,j] = sum_k P[k,i]*P[k,j] via V_WMMA_F32_16X16X4_F32.
// One wave (32 threads) per 16x16 output tile; 7x7 = 49 tiles; K = 896 = 224 x 4.
// A (16x4 f32): lane = m + 16*half, vgpr v holds K = v + 2*half, M = m.
// B (4x16 f32): symmetric, N = m.  C/D: vgpr r holds M = r + 8*half, N = m.
__global__ void __launch_bounds__(32) k_gram(float* __restrict__ ws) {
    const float* Pp = ws + WS_P;
    float* M = ws + WS_M;
    int ti = blockIdx.x / 7;
    int tj = blockIdx.x % 7;
    int lane = threadIdx.x;
    int half = lane >> 4;     // 0: K pair {0,1}; 1: K pair {2,3}
    int m = lane & 15;
    const float* colA = Pp + ti * 16 + m;
    const float* colB = Pp + tj * 16 + m;
    v8f acc = {};
    for (int k0 = 0; k0 < KK; k0 += 4) {
        int kA = k0 + 2 * half;
        v2f a, b;
        a.x = colA[(kA + 0) * NP];
        a.y = colA[(kA + 1) * NP];
        b.x = colB[(kA + 0) * NP];
        b.y = colB[(kA + 1) * NP];
        acc = __builtin_amdgcn_wmma_f32_16x16x4_f32(
            false, a, false, b, (short)0, acc, false, false);
    }
    int rowBase = ti * 16 + 8 * half;
#pragma unroll
    for (int r = 0; r < 8; ++r) {
        M[(rowBase + r) * NP + tj * 16 + m] = acc[r];
    }
}

// For each (i,j) pair: mask, eq (early-exit), L1/L2 selection, block-reduce
// sum and count, atomic accumulate.
__global__ void k_kl(const int* __restrict__ px, const int* __restrict__ py,
                     float* __restrict__ ws) {
    __shared__ float ssum[256];
    __shared__ float scnt[256];
    int e = blockIdx.x * blockDim.x + threadIdx.x;
    float lsum = 0.0f, lcnt = 0.0f;
    if (e < NPTS * NPTS) {
        int i = e / NPTS;
        int j = e - i * NPTS;
        if (i != j && px[i] != px[j] && py[i] != py[j]) {
            lcnt = 1.0f;
            bool eq = true;
            const float* Gi = ws + WS_G + i;
            const float* Gj = ws + WS_G + j;
            for (int k = 0; k < KK; ++k) {
                if (Gi[k * NP] != Gj[k * NP]) { eq = false; break; }
            }
            float cross = ws[WS_M + i * NP + j];   // symmetric
            float Dij = (ws[WS_T + j] - cross) * (1.0f / (float)NN);
            float Dji = (ws[WS_T + i] - cross) * (1.0f / (float)NN);
            if (eq) {
                lsum = Dij + Dji;
            } else {
                lsum = (float)(NN * CC) *
                       (fmaxf(2.0f - Dij, 0.0f) + fmaxf(2.0f - Dji, 0.0f));
            }
        }
    }
    ssum[threadIdx.x] = lsum;
    scnt[threadIdx.x] = lcnt;
    __syncthreads();
    for (int s = 128; s > 0; s >>= 1) {
        if ((int)threadIdx.x < s) {
            ssum[threadIdx.x] += ssum[threadIdx.x + s];
            scnt[threadIdx.x] += scnt[threadIdx.x + s];
        }
        __syncthreads();
    }
    if (threadIdx.x == 0) {
        atomicAdd(&ws[1], ssum[0]);
        atomicAdd(&ws[2], scnt[0]);
    }
}

__global__ void k_final(const float* __restrict__ ws, float* __restrict__ out) {
    if (threadIdx.x == 0) {
        float bg = ws[0] * (1.0f / (float)BG_N);
        float kl = ws[1] / ws[2];
        out[0] = bg + kl;
    }
}

extern "C" void kernel_launch(void* const* d_in, const int* in_sizes, int n_in,
                              void* d_out, int out_size, void* d_ws, size_t ws_size,
                              hipStream_t stream) {
    const float* inputs  = (const float*)d_in[0];
    const float* targets = (const float*)d_in[1];
    const int*   px      = (const int*)d_in[2];
    const int*   py      = (const int*)d_in[3];
    float* out = (float*)d_out;
    float* ws  = (float*)d_ws;
    (void)in_sizes; (void)n_in; (void)out_size; (void)ws_size;

    k_init<<<1, 32, 0, stream>>>(ws);
    k_bg<<<1536, 256, 0, stream>>>((const float4*)inputs, (const float4*)targets, ws);
    k_gather<<<(KK * NP + 255) / 256, 256, 0, stream>>>(inputs, targets, px, py, ws);
    k_term1<<<1, 128, 0, stream>>>(ws);
    k_gram<<<49, 32, 0, stream>>>(ws);
    k_kl<<<(NPTS * NPTS + 255) / 256, 256, 0, stream>>>(px, py, ws);
    k_final<<<1, 32, 0, stream>>>(ws, out);
}